// LadderCell_57913339019727
// MI455X (gfx1250) — compile-verified
//
#include <hip/hip_runtime.h>

typedef __attribute__((ext_vector_type(2))) float v2f;
typedef __attribute__((ext_vector_type(8))) float v8f;

#define B_ 128
#define T_ 256
#define D_ 256
#define U_ 512
#define O_ 512

#define NBLK 32
#define NTHR 256   // 8 waves of 32 -> 256 waves total, one 16x16 tile per wave

// ---------------------------------------------------------------------------
// fp32 WMMA: D(16x16) = A(16x4) * B(4x16) + C   (wave32, EXEC must be all 1s)
// ---------------------------------------------------------------------------
__device__ __forceinline__ v8f wmma_acc(v8f c, v2f a, v2f b) {
  return __builtin_amdgcn_wmma_f32_16x16x4_f32(
      /*neg_a=*/false, a, /*neg_b=*/false, b,
      /*c_mod=*/(short)0, c, /*reuse_a=*/false, /*reuse_b=*/false);
}

// C-tile += A[row0:+16, 0:K] * Bm[0:K, col0:+16]   (row-major, strides lda/ldb)
// A layout per ISA: lanes 0-15 hold M=lane, K={0,1} in v0/v1; lanes 16-31 K={2,3}.
// B layout (row-striped): lanes hold N, halves hold K pairs {0,1}/{2,3}.
__device__ __forceinline__ v8f gemm_tile(v8f c, const float* __restrict__ A, int lda,
                                         const float* __restrict__ Bm, int ldb,
                                         int row0, int col0, int K, int lane) {
  const int mi   = lane & 15;
  const int kOff = (lane >> 4) << 1;          // 0 or 2
  const float* aP = A + (size_t)(row0 + mi) * lda + kOff;
  const float* bP = Bm + (size_t)kOff * ldb + col0 + mi;
#pragma unroll 4
  for (int k = 0; k < K; k += 4) {
    v2f a = *(const v2f*)(aP + k);            // contiguous pair -> b64 load
    v2f b;
    b.x = bP[(size_t)k * ldb];
    b.y = bP[(size_t)k * ldb + ldb];
    c = wmma_acc(c, a, b);
  }
  return c;
}

// Same, but A element = inp[row,k] + mbuf[row,k]   (u = inp_t + m, fused)
__device__ __forceinline__ v8f gemm_tile_u(v8f c, const float* __restrict__ inp, int ldin,
                                           const float* __restrict__ mbuf,
                                           const float* __restrict__ Bm, int ldb,
                                           int row0, int col0, int K, int lane) {
  const int mi   = lane & 15;
  const int kOff = (lane >> 4) << 1;
  const float* iP = inp + (size_t)(row0 + mi) * ldin + kOff;
  const float* mP = mbuf + (size_t)(row0 + mi) * D_ + kOff;
  const float* bP = Bm + (size_t)kOff * ldb + col0 + mi;
#pragma unroll 4
  for (int k = 0; k < K; k += 4) {
    v2f ia = *(const v2f*)(iP + k);
    v2f ma = *(const v2f*)(mP + k);
    v2f a; a.x = ia.x + ma.x; a.y = ia.y + ma.y;
    v2f b;
    b.x = bP[(size_t)k * ldb];
    b.y = bP[(size_t)k * ldb + ldb];
    c = wmma_acc(c, a, b);
  }
  return c;
}

// C/D layout: VGPR r -> M = row0 + r (+8 for lanes 16-31), N = col0 + (lane&15)
__device__ __forceinline__ void store_tile(float* __restrict__ dst, int ld,
                                           int row0, int col0, v8f c, int lane) {
  const int r0  = row0 + ((lane >> 4) << 3);
  const int col = col0 + (lane & 15);
#pragma unroll
  for (int r = 0; r < 8; ++r) dst[(size_t)(r0 + r) * ld + col] = c[r];
}

// Grid-wide barrier via monotone counter in global memory (zeroed per launch).
__device__ __forceinline__ void grid_barrier(unsigned* cnt, unsigned gen) {
  __syncthreads();
  if (threadIdx.x == 0) {
    __threadfence();                               // release our writes
    atomicAdd(cnt, 1u);
    const unsigned target = gen * NBLK;
    while (*(volatile unsigned*)cnt < target) { __builtin_amdgcn_s_sleep(1); }
  }
  __syncthreads();
  __threadfence();                                 // acquire others' writes
}

__global__ __launch_bounds__(NTHR, 1)
void ladder_kernel(const float* __restrict__ inp,   // [B,T,D]
                   const float* __restrict__ wfm,   // [D]
                   const float* __restrict__ WyhT,  // [U,U]
                   const float* __restrict__ WyxT,  // [O,U]
                   const float* __restrict__ WyfT,  // [D,U]
                   const float* __restrict__ by,    // [U]
                   const float* __restrict__ AT,    // [O,O]
                   const float* __restrict__ BT,    // [D,O]
                   const float* __restrict__ CT,    // [O,D]
                   float* __restrict__ out,         // [B,T,U]
                   float* __restrict__ ws) {
  unsigned* cnt = (unsigned*)ws;
  // zeroed region: cnt pad + xbuf0 + hbuf0 + mbuf
  float* xbuf0 = ws + 64;
  float* hbuf0 = xbuf0 + B_ * O_;
  float* mbuf  = hbuf0 + B_ * U_;
  float* xbuf1 = mbuf  + B_ * D_;
  float* hbuf1 = xbuf1 + B_ * O_;
  float* fbuf  = hbuf1 + B_ * U_;

  const int lane = threadIdx.x & 31;
  const int wave = (blockIdx.x << 3) + (threadIdx.x >> 5);   // 0..255

  // x/h tiles: 8 row-tiles x 32 col-tiles over [128,512]
  const int rowX = (wave >> 5) << 4;
  const int colX = (wave & 31) << 4;
  // m tiles (waves 0..127): 8 x 16 over [128,256]
  const int rowM = (wave >> 4) << 4;
  const int colM = (wave & 15) << 4;

  const float byv = by[colX + (lane & 15)];
  const float wfv = (wave < 128) ? wfm[colM + (lane & 15)] : 0.0f;

  float* xc = xbuf0; float* xn = xbuf1;   // x ping-pong (xc starts zeroed)
  float* hc = hbuf0; float* hn = hbuf1;   // h ping-pong (hc starts zeroed)

  unsigned gen = 0;
  for (int t = 0; t < T_; ++t) {
    // ---- Phase A: x_new = x@AT + (inp_t + m)@BT -------------------------
    v8f cx = {};
    cx = gemm_tile  (cx, xc, O_, AT, O_, rowX, colX, O_, lane);
    cx = gemm_tile_u(cx, inp + (size_t)t * D_, T_ * D_, mbuf, BT, O_,
                     rowX, colX, D_, lane);
    store_tile(xn, O_, rowX, colX, cx, lane);

    ++gen; grid_barrier(cnt, gen);

    // ---- Phase B: hacc = h@WyhT + x_new@WyxT  (kept in regs);
    //               waves<128 also: m = x_new@CT, f = wfm*m ----------------
    v8f hacc = {};
    hacc = gemm_tile(hacc, hc, U_, WyhT, U_, rowX, colX, U_, lane);
    hacc = gemm_tile(hacc, xn, O_, WyxT, U_, rowX, colX, O_, lane);
    if (wave < 128) {                       // wave-uniform: EXEC stays full
      v8f cm = {};
      cm = gemm_tile(cm, xn, O_, CT, D_, rowM, colM, O_, lane);
      store_tile(mbuf, D_, rowM, colM, cm, lane);
      v8f cf;
#pragma unroll
      for (int r = 0; r < 8; ++r) cf[r] = wfv * cm[r];
      store_tile(fbuf, D_, rowM, colM, cf, lane);
    }

    ++gen; grid_barrier(cnt, gen);

    // ---- Phase C: h = tanh(hacc + f@WyfT + by) --------------------------
    hacc = gemm_tile(hacc, fbuf, D_, WyfT, U_, rowX, colX, D_, lane);
#pragma unroll
    for (int r = 0; r < 8; ++r) hacc[r] = tanhf(hacc[r] + byv);
    store_tile(hn, U_, rowX, colX, hacc, lane);
    {
      const int r0  = rowX + ((lane >> 4) << 3);
      const int col = colX + (lane & 15);
#pragma unroll
      for (int r = 0; r < 8; ++r)
        out[(size_t)(r0 + r) * ((size_t)T_ * U_) + (size_t)t * U_ + col] = hacc[r];
    }

    // swap ping-pong; no barrier needed before next Phase A (hazard-checked)
    float* tp = xc; xc = xn; xn = tp;
    tp = hc; hc = hn; hn = tp;
  }
}

extern "C" void kernel_launch(void* const* d_in, const int* in_sizes, int n_in,
                              void* d_out, int out_size, void* d_ws, size_t ws_size,
                              hipStream_t stream) {
  const float* inp  = (const float*)d_in[0];
  const float* wfm  = (const float*)d_in[1];
  const float* WyhT = (const float*)d_in[2];
  const float* WyxT = (const float*)d_in[3];
  const float* WyfT = (const float*)d_in[4];
  const float* by   = (const float*)d_in[5];
  const float* AT   = (const float*)d_in[6];
  const float* BT   = (const float*)d_in[7];
  const float* CT   = (const float*)d_in[8];

  // zero barrier counter + x0 + h0 + m0 (graph-capturable)
  const size_t zero_bytes =
      (size_t)(64 + B_ * O_ + B_ * U_ + B_ * D_) * sizeof(float);
  hipMemsetAsync(d_ws, 0, zero_bytes, stream);

  ladder_kernel<<<NBLK, NTHR, 0, stream>>>(inp, wfm, WyhT, WyxT, WyfT, by,
                                           AT, BT, CT,
                                           (float*)d_out, (float*)d_ws);
}